// OLMoE1B7B0924Synapse_37804302139863
// MI455X (gfx1250) — compile-verified
//
#include <hip/hip_runtime.h>
#include <cmath>

namespace {
constexpr int S = 512, D = 2048, H = 16, HD = 128, E = 64, TOPK = 8, M = 1024;
constexpr float EPS = 1e-5f;
constexpr float SCALE = 0.08838834764831845f;
constexpr int SLD = M + 4;  // padded LDS stride for MoE act tile
constexpr int KC = 16;      // GEMM K-chunk staged in LDS
constexpr int ALD = 20;     // padded LDS row stride (80B: 16B-aligned, bank-spread)
}

typedef float v2f __attribute__((ext_vector_type(2)));
typedef float v8f __attribute__((ext_vector_type(8)));
typedef int gv4i __attribute__((vector_size(16)));  // matches builtin prototype

__device__ __forceinline__ v8f v8zero() {
  v8f z = {0.f, 0.f, 0.f, 0.f, 0.f, 0.f, 0.f, 0.f};
  return z;
}

__device__ __forceinline__ v8f wmma4(v2f a, v2f b, v8f c) {
#if defined(__HIP_DEVICE_COMPILE__)
  // D = A(16x4) * B(4x16) + C(16x16), all f32.
  return __builtin_amdgcn_wmma_f32_16x16x4_f32(false, a, false, b, (short)0, c,
                                               false, false);
#else
  (void)a; (void)b;
  return c;
#endif
}

// --------------------------- async global->LDS copy (16B per lane) ----------
#if defined(__HIP_DEVICE_COMPILE__) && \
    __has_builtin(__builtin_amdgcn_global_load_async_to_lds_b128)
#define HAVE_ASYNC_LDS 1
#else
#define HAVE_ASYNC_LDS 0
#endif

__device__ __forceinline__ void async_copy16(const float* g, float* l) {
#if HAVE_ASYNC_LDS
  __builtin_amdgcn_global_load_async_to_lds_b128(
      (__attribute__((address_space(1))) gv4i*)g,
      (__attribute__((address_space(3))) gv4i*)l, 0, 0);
#else
  *(float4*)l = *(const float4*)g;  // synchronous fallback
#endif
}

__device__ __forceinline__ void wait_async_3() {
#if HAVE_ASYNC_LDS
#if __has_builtin(__builtin_amdgcn_s_wait_asynccnt)
  __builtin_amdgcn_s_wait_asynccnt(3);
#else
  asm volatile("s_wait_asynccnt 0x3" ::: "memory");
#endif
#endif
}

__device__ __forceinline__ void wait_async_0() {
#if HAVE_ASYNC_LDS
#if __has_builtin(__builtin_amdgcn_s_wait_asynccnt)
  __builtin_amdgcn_s_wait_asynccnt(0);
#else
  asm volatile("s_wait_asynccnt 0x0" ::: "memory");
#endif
#endif
}

// ---------------------------------------------------------------- RMSNorm ---
__global__ __launch_bounds__(256) void rms_kernel(const float* __restrict__ x,
                                                  const float* __restrict__ w,
                                                  float* __restrict__ out) {
  const int t = blockIdx.x;
  const float* xr = x + (size_t)t * D;
  float ss = 0.f;
  for (int i = threadIdx.x; i < D; i += 256) {
    float v = xr[i];
    ss = fmaf(v, v, ss);
  }
#pragma unroll
  for (int o = 16; o > 0; o >>= 1) ss += __shfl_xor(ss, o, 32);
  __shared__ float red[8];
  __shared__ float rnorm;
  const int wid = threadIdx.x >> 5;
  if ((threadIdx.x & 31) == 0) red[wid] = ss;
  __syncthreads();
  if (threadIdx.x == 0) {
    float s = 0.f;
    for (int i = 0; i < 8; ++i) s += red[i];
    rnorm = rsqrtf(s / (float)D + EPS);
  }
  __syncthreads();
  const float rn = rnorm;
  for (int i = threadIdx.x; i < D; i += 256)
    out[(size_t)t * D + i] = xr[i] * rn * w[i];
}

// ----------------------------------------------- Q/K RMSNorm + RoPE (inplace)
__global__ __launch_bounds__(256) void qknorm_rope_kernel(
    float* __restrict__ q, float* __restrict__ k,
    const float* __restrict__ wqn, const float* __restrict__ wkn) {
  const int t = blockIdx.x;
  float* qr = q + (size_t)t * D;
  float* kr = k + (size_t)t * D;
  float sq = 0.f, sk = 0.f;
  for (int i = threadIdx.x; i < D; i += 256) {
    float a = qr[i];
    sq = fmaf(a, a, sq);
    float b = kr[i];
    sk = fmaf(b, b, sk);
  }
#pragma unroll
  for (int o = 16; o > 0; o >>= 1) {
    sq += __shfl_xor(sq, o, 32);
    sk += __shfl_xor(sk, o, 32);
  }
  __shared__ float rq8[8], rk8[8];
  __shared__ float rqs, rks;
  const int wid = threadIdx.x >> 5;
  if ((threadIdx.x & 31) == 0) {
    rq8[wid] = sq;
    rk8[wid] = sk;
  }
  __syncthreads();
  if (threadIdx.x == 0) {
    float a = 0.f, b = 0.f;
    for (int i = 0; i < 8; ++i) {
      a += rq8[i];
      b += rk8[i];
    }
    rqs = rsqrtf(a / (float)D + EPS);
    rks = rsqrtf(b / (float)D + EPS);
  }
  __syncthreads();
  const float nq = rqs, nk = rks;
  const float pos = (float)t;
  for (int idx = threadIdx.x; idx < H * (HD / 2); idx += 256) {
    const int h = idx >> 6;
    const int i = idx & 63;
    const int base = h * HD;
    const float inv_freq = __powf(10000.0f, -(float)i / 64.0f);
    const float ang = pos * inv_freq;
    const float c = __cosf(ang), s = __sinf(ang);
    float q1 = qr[base + i] * nq * wqn[base + i];
    float q2 = qr[base + i + 64] * nq * wqn[base + i + 64];
    qr[base + i] = q1 * c - q2 * s;
    qr[base + i + 64] = q1 * s + q2 * c;
    float k1 = kr[base + i] * nk * wkn[base + i];
    float k2 = kr[base + i + 64] * nk * wkn[base + i + 64];
    kr[base + i] = k1 * c - k2 * s;
    kr[base + i + 64] = k1 * s + k2 * c;
  }
}

// ---------- C[M,N] = A[M,K] @ W[N,K]^T (+R), async double-buffered in LDS ---
__global__ __launch_bounds__(256) void gemm_wmma_async_kernel(
    const float* __restrict__ A, const float* __restrict__ W,
    const float* __restrict__ resid, float* __restrict__ C, int K, int N) {
  __shared__ __align__(16) float Ab[2][64 * ALD];
  __shared__ __align__(16) float Bb[2][128 * ALD];
  const int bm = blockIdx.x * 64;
  const int bn = blockIdx.y * 128;
  const int tid = threadIdx.x;
  const int lane = tid & 31;
  const int wid = tid >> 5;
  const int wm = (wid >> 2) * 32;  // 0 or 32
  const int wn = (wid & 3) * 32;   // 0..96
  const int m = lane & 15;
  const int kh = (lane >> 4) * 2;

  // staging role: each thread moves 3x16B per chunk
  const int srow = tid >> 2;         // 0..63
  const int sseg = (tid & 3) * 4;    // 0,4,8,12 (floats)
  const float* ag = A + (size_t)(bm + srow) * K + sseg;
  const float* bg0 = W + (size_t)(bn + srow) * K + sseg;
  const float* bg1 = W + (size_t)(bn + 64 + srow) * K + sseg;
  const int lofsA = srow * ALD + sseg;
  const int lofsB0 = srow * ALD + sseg;
  const int lofsB1 = (64 + srow) * ALD + sseg;

  v8f c00 = v8zero(), c01 = v8zero(), c10 = v8zero(), c11 = v8zero();

  const int NC = K / KC;
  // prologue: fill buffer 0
  {
    async_copy16(ag, &Ab[0][lofsA]);
    async_copy16(bg0, &Bb[0][lofsB0]);
    async_copy16(bg1, &Bb[0][lofsB1]);
  }
  for (int kc = 0; kc < NC; ++kc) {
    const int b = kc & 1;
    if (kc + 1 < NC) {
      const int off = (kc + 1) * KC;
      async_copy16(ag + off, &Ab[b ^ 1][lofsA]);
      async_copy16(bg0 + off, &Bb[b ^ 1][lofsB0]);
      async_copy16(bg1 + off, &Bb[b ^ 1][lofsB1]);
      wait_async_3();  // chunk kc landed (in-order completion)
    } else {
      wait_async_0();
    }
    __syncthreads();  // publish chunk kc to all waves

    const float* As0 = &Ab[b][(wm + m) * ALD + kh];
    const float* As1 = As0 + 16 * ALD;
    const float* Bs0 = &Bb[b][(wn + m) * ALD + kh];
    const float* Bs1 = Bs0 + 16 * ALD;
#pragma unroll
    for (int k = 0; k < KC; k += 4) {
      v2f a0 = *(const v2f*)(As0 + k);
      v2f a1 = *(const v2f*)(As1 + k);
      v2f b0 = *(const v2f*)(Bs0 + k);
      v2f b1 = *(const v2f*)(Bs1 + k);
      c00 = wmma4(a0, b0, c00);
      c01 = wmma4(a0, b1, c01);
      c10 = wmma4(a1, b0, c10);
      c11 = wmma4(a1, b1, c11);
    }
    __syncthreads();  // all waves done reading buffer b before it is refilled
  }

  const int col = lane & 15;
  const int rsel = (lane >> 4) * 8;
#pragma unroll
  for (int g = 0; g < 8; ++g) {
    const int r0 = bm + wm + g + rsel;
    const int r1 = r0 + 16;
    const int cc0 = bn + wn + col;
    const int cc1 = cc0 + 16;
    float o00 = c00[g], o01 = c01[g], o10 = c10[g], o11 = c11[g];
    if (resid) {
      o00 += resid[(size_t)r0 * N + cc0];
      o01 += resid[(size_t)r0 * N + cc1];
      o10 += resid[(size_t)r1 * N + cc0];
      o11 += resid[(size_t)r1 * N + cc1];
    }
    C[(size_t)r0 * N + cc0] = o00;
    C[(size_t)r0 * N + cc1] = o01;
    C[(size_t)r1 * N + cc0] = o10;
    C[(size_t)r1 * N + cc1] = o11;
  }
}

// ------------------------------------------------ flash attention (causal) --
__global__ __launch_bounds__(128) void attn_kernel(const float* __restrict__ Q,
                                                   const float* __restrict__ Kmat,
                                                   const float* __restrict__ V,
                                                   float* __restrict__ ctx) {
  const int h = blockIdx.y;
  const int wid = threadIdx.x >> 5;
  const int lane = threadIdx.x & 31;
  const int q0 = blockIdx.x * 64 + wid * 16;  // this wave's 16 query rows
  const int m = lane & 15;
  const int kh = (lane >> 4) * 2;
  const int rsel = (lane >> 4) * 8;
  __shared__ float pl[4][16 * 17];  // per-wave P tile, padded stride
  float* pw = pl[wid];

  v2f qreg[32];
  {
    const float* qrow = Q + (size_t)(q0 + m) * D + h * HD + kh;
#pragma unroll
    for (int s = 0; s < 32; ++s) qreg[s] = *(const v2f*)(qrow + s * 4);
  }
  float mrow[8], lrow[8];
  v8f oacc[8];
#pragma unroll
  for (int g = 0; g < 8; ++g) {
    mrow[g] = -1e30f;
    lrow[g] = 0.f;
    oacc[g] = v8zero();
  }

  const int ktmax = q0 >> 4;  // inclusive (q0 is a multiple of 16)
  for (int kt = 0; kt <= ktmax; ++kt) {
    v8f sc = v8zero();
    const float* kb = Kmat + (size_t)(kt * 16 + m) * D + h * HD + kh;
#pragma unroll
    for (int s = 0; s < 32; ++s) {
      v2f b = *(const v2f*)(kb + s * 4);
      sc = wmma4(qreg[s], b, sc);
    }
    const int krow = kt * 16 + m;  // key index = this lane's column
#pragma unroll
    for (int g = 0; g < 8; ++g) {
      const int qrow = q0 + g + rsel;
      float v = sc[g] * SCALE;
      if (krow > qrow) v = -1e30f;  // causal mask
      float rmax = v;
#pragma unroll
      for (int o = 1; o < 16; o <<= 1) rmax = fmaxf(rmax, __shfl_xor(rmax, o, 16));
      const float mnew = fmaxf(mrow[g], rmax);
      const float p = __expf(v - mnew);
      float rsum = p;
#pragma unroll
      for (int o = 1; o < 16; o <<= 1) rsum += __shfl_xor(rsum, o, 16);
      const float alpha = __expf(mrow[g] - mnew);
      mrow[g] = mnew;
      lrow[g] = lrow[g] * alpha + rsum;
#pragma unroll
      for (int nt = 0; nt < 8; ++nt) oacc[nt][g] *= alpha;
      sc[g] = p;
    }
#pragma unroll
    for (int g = 0; g < 8; ++g) pw[(g + rsel) * 17 + m] = sc[g];
#pragma unroll
    for (int nt = 0; nt < 8; ++nt) {
      v8f acc = oacc[nt];
#pragma unroll
      for (int st = 0; st < 4; ++st) {
        const int kk = st * 4 + kh;
        v2f a;
        a.x = pw[m * 17 + kk];
        a.y = pw[m * 17 + kk + 1];
        const float* vb = V + (size_t)(kt * 16 + kk) * D + h * HD + nt * 16 + m;
        v2f b;
        b.x = vb[0];
        b.y = vb[D];
        acc = wmma4(a, b, acc);
      }
      oacc[nt] = acc;
    }
  }
#pragma unroll
  for (int g = 0; g < 8; ++g) {
    const float inv = 1.0f / lrow[g];
    const int row = q0 + g + rsel;
#pragma unroll
    for (int nt = 0; nt < 8; ++nt)
      ctx[(size_t)row * D + h * HD + nt * 16 + m] = oacc[nt][g] * inv;
  }
}

// ---------------------------------------- router: softmax + top-8 + scatter -
__global__ __launch_bounds__(256) void router_kernel(
    const float* __restrict__ x3, const float* __restrict__ wr,
    int* __restrict__ cnt, int* __restrict__ toks, float* __restrict__ gates) {
  const int t = blockIdx.x;
  __shared__ float xs[D];
  for (int i = threadIdx.x; i < D; i += 256) xs[i] = x3[(size_t)t * D + i];
  __syncthreads();
  __shared__ float lg[E];
  if (threadIdx.x < E) {
    const float* w = wr + (size_t)threadIdx.x * D;
    float s = 0.f;
    for (int i = 0; i < D; ++i) s = fmaf(xs[i], w[i], s);
    lg[threadIdx.x] = s;
  }
  __syncthreads();
  if (threadIdx.x == 0) {
    float mx = -1e30f;
    for (int e = 0; e < E; ++e) mx = fmaxf(mx, lg[e]);
    float pr[E];
    float sum = 0.f;
    for (int e = 0; e < E; ++e) {
      pr[e] = __expf(lg[e] - mx);
      sum += pr[e];
    }
    const float inv = 1.0f / sum;
    for (int j = 0; j < TOPK; ++j) {
      int best = 0;
      float bv = -1.f;
      for (int e = 0; e < E; ++e)
        if (pr[e] > bv) {
          bv = pr[e];
          best = e;
        }
      pr[best] = -2.f;
      const int pos = atomicAdd(&cnt[best], 1);
      toks[best * S + pos] = t;
      gates[best * S + pos] = bv * inv;
    }
  }
}

// --------------------------------------------------- sparse MoE expert FFN --
__global__ __launch_bounds__(256) void moe_kernel(
    const float* __restrict__ x3, const float* __restrict__ wgu,
    const float* __restrict__ wdn, const int* __restrict__ cnt,
    const int* __restrict__ toks, const float* __restrict__ gates,
    float* __restrict__ out) {
  const int e = blockIdx.x;
  const int tile = blockIdx.y;
  const int nt = cnt[e];
  if (tile * 16 >= nt) return;
  __shared__ int stok[16];
  __shared__ float sgate[16];
  __shared__ float sact[16 * SLD];  // 16 x 1024 act tile, padded stride
  if (threadIdx.x < 16) {
    const int idx = tile * 16 + threadIdx.x;
    if (idx < nt) {
      stok[threadIdx.x] = toks[e * S + idx];
      sgate[threadIdx.x] = gates[e * S + idx];
    } else {
      stok[threadIdx.x] = 0;
      sgate[threadIdx.x] = 0.f;  // padded row contributes nothing
    }
  }
  __syncthreads();
  const int lane = threadIdx.x & 31;
  const int wid = threadIdx.x >> 5;
  const int m = lane & 15;
  const int kh = (lane >> 4) * 2;
  const int rsel = (lane >> 4) * 8;
  const float* xrow = x3 + (size_t)stok[m] * D + kh;
  const float* wg = wgu + (size_t)e * (2 * M) * D;

  // GEMM1: gate/up pairs, K = D. 64 col tiles -> 8 per wave.
  for (int jt = wid; jt < M / 16; jt += 8) {
    v8f cg = v8zero(), cu = v8zero();
    const float* wgr = wg + (size_t)(jt * 16 + m) * D + kh;
    const float* wur = wg + (size_t)(M + jt * 16 + m) * D + kh;
#pragma unroll 4
    for (int k = 0; k < D; k += 4) {
      v2f a = *(const v2f*)(xrow + k);
      v2f bg = *(const v2f*)(wgr + k);
      v2f bu = *(const v2f*)(wur + k);
      cg = wmma4(a, bg, cg);
      cu = wmma4(a, bu, cu);
    }
#pragma unroll
    for (int g = 0; g < 8; ++g) {
      const float gv = cg[g];
      const float act = (gv / (1.f + __expf(-gv))) * cu[g] * sgate[g + rsel];
      sact[(g + rsel) * SLD + jt * 16 + m] = act;
    }
  }
  __syncthreads();

  // GEMM2: down projection, K = M. 128 d tiles -> 16 per wave.
  const float* wd = wdn + (size_t)e * D * M;
  for (int dt = wid; dt < D / 16; dt += 8) {
    v8f acc = v8zero();
    const float* wdr = wd + (size_t)(dt * 16 + m) * M + kh;
#pragma unroll 4
    for (int k = 0; k < M; k += 4) {
      v2f a;
      a.x = sact[m * SLD + k + kh];
      a.y = sact[m * SLD + k + kh + 1];
      v2f b = *(const v2f*)(wdr + k);
      acc = wmma4(a, b, acc);
    }
#pragma unroll
    for (int g = 0; g < 8; ++g) {
      const int row = stok[g + rsel];
      unsafeAtomicAdd(&out[(size_t)row * D + dt * 16 + m], acc[g]);
    }
  }
}

// ---------------------------------------------------------------------------
extern "C" void kernel_launch(void* const* d_in, const int* in_sizes, int n_in,
                              void* d_out, int out_size, void* d_ws,
                              size_t ws_size, hipStream_t stream) {
  (void)in_sizes; (void)n_in; (void)out_size; (void)ws_size;
  const float* x = (const float*)d_in[0];
  const float* w_in_ln = (const float*)d_in[1];
  const float* w_q = (const float*)d_in[2];
  const float* w_k = (const float*)d_in[3];
  const float* w_v = (const float*)d_in[4];
  const float* w_qn = (const float*)d_in[5];
  const float* w_kn = (const float*)d_in[6];
  const float* w_o = (const float*)d_in[7];
  const float* w_post = (const float*)d_in[8];
  const float* w_rt = (const float*)d_in[9];
  const float* w_gu = (const float*)d_in[10];
  const float* w_dn = (const float*)d_in[11];
  float* out = (float*)d_out;

  float* ws = (float*)d_ws;
  const size_t TD = (size_t)S * D;
  float* xn = ws;
  float* q = ws + TD;
  float* k = ws + 2 * TD;
  float* v = ws + 3 * TD;
  float* ctx = ws + 4 * TD;
  float* x3 = ws + 5 * TD;
  int* cnt = (int*)(ws + 6 * TD);
  int* toks = cnt + 64;
  float* gatesbuf = (float*)(toks + E * S);

  rms_kernel<<<S, 256, 0, stream>>>(x, w_in_ln, xn);

  dim3 gg(S / 64, D / 128);
  gemm_wmma_async_kernel<<<gg, 256, 0, stream>>>(xn, w_q, nullptr, q, D, D);
  gemm_wmma_async_kernel<<<gg, 256, 0, stream>>>(xn, w_k, nullptr, k, D, D);
  gemm_wmma_async_kernel<<<gg, 256, 0, stream>>>(xn, w_v, nullptr, v, D, D);

  qknorm_rope_kernel<<<S, 256, 0, stream>>>(q, k, w_qn, w_kn);

  attn_kernel<<<dim3(S / 64, H), 128, 0, stream>>>(q, k, v, ctx);

  // x2 = x + ctx @ w_o^T  (written to out; MoE adds on top)
  gemm_wmma_async_kernel<<<gg, 256, 0, stream>>>(ctx, w_o, x, out, D, D);

  rms_kernel<<<S, 256, 0, stream>>>(out, w_post, x3);

  (void)hipMemsetAsync(cnt, 0, E * sizeof(int), stream);
  router_kernel<<<S, 256, 0, stream>>>(x3, w_rt, cnt, toks, gatesbuf);

  moe_kernel<<<dim3(E, S / 16), 256, 0, stream>>>(x3, w_gu, w_dn, cnt, toks,
                                                  gatesbuf, out);
}